// GATLinkPredictor_49100066128264
// MI455X (gfx1250) — compile-verified
//
#include <hip/hip_runtime.h>
#include <hip/hip_bf16.h>
#include <math.h>

typedef __attribute__((ext_vector_type(2))) float v2f;
typedef __attribute__((ext_vector_type(8))) float v8f;

#define NEG_SLOPE 0.2f

__device__ __forceinline__ float lrelu(float v) { return v > 0.f ? v : NEG_SLOPE * v; }

// order-preserving float -> uint encoding (for atomicMax on floats)
__device__ __forceinline__ unsigned enc_f(float f) {
    unsigned u = __float_as_uint(f);
    return (u & 0x80000000u) ? ~u : (u | 0x80000000u);
}
__device__ __forceinline__ float dec_f(unsigned k) {
    unsigned u = (k & 0x80000000u) ? (k & 0x7fffffffu) : ~k;
    return __uint_as_float(u);
}

__global__ void fill_u32_kernel(unsigned* __restrict__ p, long long n, unsigned val) {
    long long i = (long long)blockIdx.x * blockDim.x + threadIdx.x;
    long long stride = (long long)gridDim.x * blockDim.x;
    for (; i < n; i += stride) p[i] = val;
}

// ---------------------------------------------------------------------------
// C[M x (NTILES*16)] = A[M x 128] * B[128 x (NTILES*16)]   (fp32, WMMA 16x16x4)
// 256 threads = 8 wave32; each block computes a 128-row band of C.
// Wave w owns NTILES accumulator tiles (tile ids w, w+8, w+16, ...).
// ---------------------------------------------------------------------------
template <int NTILES>
__global__ __launch_bounds__(256) void gemm128_wmma_kernel(
    const float* __restrict__ A, const float* __restrict__ B,
    float* __restrict__ C, int M)
{
    constexpr int NCOLS = NTILES * 16;
    __shared__ float ldsA[128 * 33];   // 128 rows x 32 K (padded)
    __shared__ float ldsB[32 * 130];   // 32 K x <=128 cols (padded)

    const int tid  = threadIdx.x;
    const int wave = tid >> 5;
    const int lane = tid & 31;
    const int l16  = lane & 15;
    const int hi   = lane >> 4;        // 0 for lanes 0-15, 1 for lanes 16-31
    const int hi2  = hi * 2;
    const int mbase = blockIdx.x * 128;

    v8f acc[NTILES] = {};

    for (int kk = 0; kk < 128; kk += 32) {
        // load A chunk: rows [mbase, mbase+128), cols [kk, kk+32)
        for (int idx = tid; idx < 128 * 32; idx += 256) {
            int r = idx >> 5, kc = idx & 31;
            int row = mbase + r;
            ldsA[r * 33 + kc] = (row < M) ? A[(long long)row * 128 + kk + kc] : 0.f;
        }
        // load B chunk: rows [kk, kk+32), all NCOLS cols
        for (int idx = tid; idx < 32 * NCOLS; idx += 256) {
            int k = idx / NCOLS, n = idx - k * NCOLS;
            ldsB[k * 130 + n] = B[(long long)(kk + k) * NCOLS + n];
        }
        __syncthreads();

#pragma unroll
        for (int t = 0; t < NTILES; ++t) {
            const int tidx = wave + 8 * t;
            const int mt = tidx / NTILES, nt = tidx - mt * NTILES;
            const float* pa = &ldsA[(mt * 16 + l16) * 33];
            const float* pb = &ldsB[nt * 16 + l16];
#pragma unroll
            for (int k4 = 0; k4 < 32; k4 += 4) {
                v2f a, b;
                // A 16x4 layout: lanes 0-15 hold K=k4+{0,1}, lanes 16-31 K=k4+{2,3}
                a[0] = pa[k4 + hi2 + 0];
                a[1] = pa[k4 + hi2 + 1];
                // B 4x16 layout: mirrored (rows = K, striped across lanes by N)
                b[0] = pb[(k4 + hi2 + 0) * 130];
                b[1] = pb[(k4 + hi2 + 1) * 130];
                acc[t] = __builtin_amdgcn_wmma_f32_16x16x4_f32(
                    false, a, false, b, (short)0, acc[t], false, false);
            }
        }
        __syncthreads();
    }

    // store: C tile layout — VGPR v holds M = v + 8*hi, N striped over lanes 0-15
#pragma unroll
    for (int t = 0; t < NTILES; ++t) {
        const int tidx = wave + 8 * t;
        const int mt = tidx / NTILES, nt = tidx - mt * NTILES;
        const int col = nt * 16 + l16;
        const int rbase = mbase + mt * 16 + hi * 8;
#pragma unroll
        for (int v = 0; v < 8; ++v) {
            int row = rbase + v;
            if (row < M) C[(long long)row * NCOLS + col] = acc[t][v];
        }
    }
}

// al_s[n,h] = sum_c hlin[n,h,c]*a_src[h,c] ; al_d likewise
__global__ void att_score_kernel(const float* __restrict__ hlin,
                                 const float* __restrict__ a_src,
                                 const float* __restrict__ a_dst,
                                 float* __restrict__ als, float* __restrict__ ald,
                                 int N, int H, int C)
{
    int i = blockIdx.x * blockDim.x + threadIdx.x;  // i = n*H + h
    if (i >= N * H) return;
    int h = i % H, n = i / H;
    const float* hp = hlin + (long long)n * (H * C) + h * C;
    const float* as = a_src + h * C;
    const float* ad = a_dst + h * C;
    float ss = 0.f, sd = 0.f;
    for (int c = 0; c < C; ++c) { float v = hp[c]; ss += v * as[c]; sd += v * ad[c]; }
    als[i] = ss; ald[i] = sd;
}

// pass 1: per-dst per-head running max of leaky_relu(al_s[src]+al_d[dst])
__global__ void edge_max_kernel(const int* __restrict__ src, const int* __restrict__ dst,
                                int E, int N, const float* __restrict__ als,
                                const float* __restrict__ ald, int Hshift,
                                unsigned* __restrict__ m)
{
    long long i = (long long)blockIdx.x * blockDim.x + threadIdx.x;
    const int H = 1 << Hshift;
    long long total = (long long)(E + N) << Hshift;
    if (i >= total) return;
    int h = (int)(i & (H - 1));
    long long e = i >> Hshift;
    int s, d;
    if (e < E) { s = src[e]; d = dst[e]; } else { s = d = (int)(e - E); }
    float val = lrelu(als[(long long)s * H + h] + ald[(long long)d * H + h]);
    atomicMax(&m[(long long)d * H + h], enc_f(val));
}

// pass 2: denom[dst,h] += exp(e - m[dst,h])
__global__ void edge_sum_kernel(const int* __restrict__ src, const int* __restrict__ dst,
                                int E, int N, const float* __restrict__ als,
                                const float* __restrict__ ald, int Hshift,
                                const unsigned* __restrict__ m, float* __restrict__ den)
{
    long long i = (long long)blockIdx.x * blockDim.x + threadIdx.x;
    const int H = 1 << Hshift;
    long long total = (long long)(E + N) << Hshift;
    if (i >= total) return;
    int h = (int)(i & (H - 1));
    long long e = i >> Hshift;
    int s, d;
    if (e < E) { s = src[e]; d = dst[e]; } else { s = d = (int)(e - E); }
    float val = lrelu(als[(long long)s * H + h] + ald[(long long)d * H + h]);
    float mv = dec_f(m[(long long)d * H + h]);
    atomicAdd(&den[(long long)d * H + h], expf(val - mv));
}

// pass 3: out[dst,f] += hlin[src,f] * alpha   (one wave32 per edge)
__global__ void edge_aggregate_kernel(const int* __restrict__ src, const int* __restrict__ dst,
                                      int E, int N, const float* __restrict__ hlin,
                                      const float* __restrict__ als, const float* __restrict__ ald,
                                      const unsigned* __restrict__ m, const float* __restrict__ den,
                                      float* __restrict__ out, int Hshift, int Cshift, int F)
{
    long long gid = (long long)blockIdx.x * blockDim.x + threadIdx.x;
    long long wv = gid >> 5;
    int lane = (int)(gid & 31);
    long long Etot = (long long)E + N;
    if (wv >= Etot) return;
    int s, d;
    if (wv < E) { s = src[wv]; d = dst[wv]; } else { s = d = (int)(wv - E); }
    const int H = 1 << Hshift;
    for (int f = lane; f < F; f += 32) {
        int h = f >> Cshift;
        float e  = lrelu(als[(long long)s * H + h] + ald[(long long)d * H + h]);
        float mv = dec_f(m[(long long)d * H + h]);
        float alpha = expf(e - mv) / (den[(long long)d * H + h] + 1e-16f);
        atomicAdd(&out[(long long)d * F + f], hlin[(long long)s * F + f] * alpha);
    }
}

__global__ void bias_act_kernel(float* __restrict__ io, const float* __restrict__ bias,
                                long long total, int F, int do_elu)
{
    long long i = (long long)blockIdx.x * blockDim.x + threadIdx.x;
    if (i >= total) return;
    float v = io[i] + bias[(int)(i & (F - 1))];
    if (do_elu) v = v > 0.f ? v : expm1f(v);
    io[i] = v;
}

// decode: out[e] = dot(z[a], z[b]) over 64 features (wave per label edge)
__global__ void decode_kernel(const float* __restrict__ z, const int* __restrict__ eli,
                              int EL, float* __restrict__ out)
{
    long long gid = (long long)blockIdx.x * blockDim.x + threadIdx.x;
    long long wv = gid >> 5;
    int lane = (int)(gid & 31);
    if (wv >= EL) return;
    int a = eli[wv];
    int b = eli[(long long)EL + wv];
    const float* za = z + (long long)a * 64;
    const float* zb = z + (long long)b * 64;
    float v = za[lane] * zb[lane] + za[lane + 32] * zb[lane + 32];
    v += __shfl_xor(v, 16);
    v += __shfl_xor(v, 8);
    v += __shfl_xor(v, 4);
    v += __shfl_xor(v, 2);
    v += __shfl_xor(v, 1);
    if (lane == 0) out[wv] = v;
}

static inline unsigned nblk(long long n, int b) { return (unsigned)((n + b - 1) / b); }

extern "C" void kernel_launch(void* const* d_in, const int* in_sizes, int n_in,
                              void* d_out, int out_size, void* d_ws, size_t ws_size,
                              hipStream_t stream)
{
    const float* x   = (const float*)d_in[0];
    const int*   ei  = (const int*)d_in[1];
    const int*   eli = (const int*)d_in[2];
    const float* W1  = (const float*)d_in[3];
    const float* as1 = (const float*)d_in[4];
    const float* ad1 = (const float*)d_in[5];
    const float* b1  = (const float*)d_in[6];
    const float* W2  = (const float*)d_in[7];
    const float* as2 = (const float*)d_in[8];
    const float* ad2 = (const float*)d_in[9];
    const float* b2  = (const float*)d_in[10];

    const int N  = in_sizes[0] / 128;
    const int E  = in_sizes[1] / 2;
    const int EL = in_sizes[2] / 2;
    const int F1 = 128, F2 = 64;
    const int H1 = 4, C1 = 32, C2 = 64;

    const int* src = ei;          // edge_index row 0
    const int* dst = ei + E;      // edge_index row 1

    // ---- workspace carve-up (256B aligned) ----
    char* ws = (char*)d_ws;
    auto al256 = [](size_t s) { return (s + 255) & ~(size_t)255; };
    size_t off = 0;
    float*    hlin1 = (float*)(ws + off);    off += al256((size_t)N * F1 * 4);
    float*    out1  = (float*)(ws + off);    off += al256((size_t)N * F1 * 4);
    float*    als1  = (float*)(ws + off);    off += al256((size_t)N * H1 * 4);
    float*    ald1  = (float*)(ws + off);    off += al256((size_t)N * H1 * 4);
    unsigned* m1    = (unsigned*)(ws + off); off += al256((size_t)N * H1 * 4);
    float*    den1  = (float*)(ws + off);    off += al256((size_t)N * H1 * 4);
    float*    als2  = (float*)(ws + off);    off += al256((size_t)N * 4);
    float*    ald2  = (float*)(ws + off);    off += al256((size_t)N * 4);
    unsigned* m2    = (unsigned*)(ws + off); off += al256((size_t)N * 4);
    float*    den2  = (float*)(ws + off);    off += al256((size_t)N * 4);
    // layer-2 big buffers alias hlin1 (dead after layer-1 aggregation)
    float* hlin2 = hlin1;
    float* out2  = (float*)((char*)hlin1 + al256((size_t)N * F2 * 4));

    const int B = 256;
    const long long Etot = (long long)E + N;

    // ===== layer 1 =====
    fill_u32_kernel<<<2048, B, 0, stream>>>((unsigned*)out1, (long long)N * F1, 0u);
    fill_u32_kernel<<<512,  B, 0, stream>>>(m1, (long long)N * H1, 0u);
    fill_u32_kernel<<<512,  B, 0, stream>>>((unsigned*)den1, (long long)N * H1, 0u);

    gemm128_wmma_kernel<8><<<nblk(N, 128), B, 0, stream>>>(x, W1, hlin1, N);

    att_score_kernel<<<nblk((long long)N * H1, B), B, 0, stream>>>(
        hlin1, as1, ad1, als1, ald1, N, H1, C1);

    edge_max_kernel<<<nblk(Etot * H1, B), B, 0, stream>>>(src, dst, E, N, als1, ald1, 2, m1);
    edge_sum_kernel<<<nblk(Etot * H1, B), B, 0, stream>>>(src, dst, E, N, als1, ald1, 2, m1, den1);
    edge_aggregate_kernel<<<nblk(Etot * 32, B), B, 0, stream>>>(
        src, dst, E, N, hlin1, als1, ald1, m1, den1, out1, 2, 5, F1);

    bias_act_kernel<<<nblk((long long)N * F1, B), B, 0, stream>>>(
        out1, b1, (long long)N * F1, F1, 1);   // + bias, ELU  -> h2 in out1

    // ===== layer 2 =====
    fill_u32_kernel<<<2048, B, 0, stream>>>((unsigned*)out2, (long long)N * F2, 0u);
    fill_u32_kernel<<<256,  B, 0, stream>>>(m2, (long long)N, 0u);
    fill_u32_kernel<<<256,  B, 0, stream>>>((unsigned*)den2, (long long)N, 0u);

    gemm128_wmma_kernel<4><<<nblk(N, 128), B, 0, stream>>>(out1, W2, hlin2, N);

    att_score_kernel<<<nblk((long long)N, B), B, 0, stream>>>(
        hlin2, as2, ad2, als2, ald2, N, 1, C2);

    edge_max_kernel<<<nblk(Etot, B), B, 0, stream>>>(src, dst, E, N, als2, ald2, 0, m2);
    edge_sum_kernel<<<nblk(Etot, B), B, 0, stream>>>(src, dst, E, N, als2, ald2, 0, m2, den2);
    edge_aggregate_kernel<<<nblk(Etot * 32, B), B, 0, stream>>>(
        src, dst, E, N, hlin2, als2, ald2, m2, den2, out2, 0, 6, F2);

    bias_act_kernel<<<nblk((long long)N * F2, B), B, 0, stream>>>(
        out2, b2, (long long)N * F2, F2, 0);   // + bias, no activation -> z

    // ===== decode =====
    decode_kernel<<<nblk((long long)EL * 32, B), B, 0, stream>>>(out2, eli, EL, (float*)d_out);
}